// MultiHeadAttention_27917287424346
// MI455X (gfx1250) — compile-verified
//
#include <hip/hip_runtime.h>

typedef __attribute__((ext_vector_type(2))) float v2f;
typedef __attribute__((ext_vector_type(4))) float v4f;
typedef __attribute__((ext_vector_type(8))) float v8f;
typedef __attribute__((ext_vector_type(4))) int   v4i;

#define NROW   16384   // t*b*s
#define DMODEL 512
#define SEQ    1024    // reshaped "seq" length (row-major no-op reshape)
#define DH     64
#define NGRP   128     // t*b*h

// ---------------------------------------------------------------------------
// CDNA5 async global->LDS copy (ASYNCcnt-tracked), guarded fallback.
// Confirmed signature: (v4i addrspace(1)*, v4i addrspace(3)*, imm, imm)
// ---------------------------------------------------------------------------
#if __has_builtin(__builtin_amdgcn_global_load_async_to_lds_b128) && \
    __has_builtin(__builtin_amdgcn_s_wait_asynccnt)
#define HAVE_ASYNC 1
#else
#define HAVE_ASYNC 0
#endif

typedef __attribute__((address_space(1))) v4i* as1_v4i;
typedef __attribute__((address_space(3))) v4i* as3_v4i;

static __device__ __forceinline__ void copy16_g2l(const float* __restrict__ g,
                                                  float* __restrict__ l) {
#if HAVE_ASYNC
  // flat->LDS offset: hardware truncates LDS-aperture flat addr to addr[31:0]
  __builtin_amdgcn_global_load_async_to_lds_b128(
      (as1_v4i)(unsigned long long)g,
      (as3_v4i)(unsigned)(unsigned long long)l, 0, 0);
#else
  const v4f t = *(const v4f*)g;
  l[0] = t[0]; l[1] = t[1]; l[2] = t[2]; l[3] = t[3];
#endif
}

template <int N>
static __device__ __forceinline__ void wait_async() {
#if HAVE_ASYNC
  __builtin_amdgcn_s_wait_asynccnt(N);
#endif
}

static __device__ __forceinline__ v8f wmma_f32(v2f a, v2f b, v8f c) {
  // V_WMMA_F32_16X16X4_F32: D = A(16x4) * B(4x16) + C(16x16), all f32.
  return __builtin_amdgcn_wmma_f32_16x16x4_f32(
      false, a, false, b, (short)0, c, false, false);
}

// ---------------------------------------------------------------------------
// Weight pre-interleave: Wt[((k>>1)*512 + n)*2 + (k&1)] = W[k*512 + n].
// Makes every GEMM B-fragment a single contiguous b64 in LDS while keeping
// async b128 tile fills contiguous in both global and LDS.
// ---------------------------------------------------------------------------
__global__ __launch_bounds__(256)
void interleave_w_kernel(const float* __restrict__ W, float* __restrict__ Wt) {
  const int idx = blockIdx.x * 256 + threadIdx.x;  // 0..131071
  const int p = idx >> 9;                          // k-pair 0..255
  const int n = idx & 511;
  v2f t;
  t[0] = W[(size_t)(2 * p) * DMODEL + n];
  t[1] = W[(size_t)(2 * p + 1) * DMODEL + n];
  *(v2f*)&Wt[((size_t)p * DMODEL + n) * 2] = t;
}

// ---------------------------------------------------------------------------
// GEMM: C[16384,512] = A[16384,512] @ W[512,512] + bias[512]
// block = 256 threads (8 waves); tile 128(M) x 64(N); BK = 16.
// Double-buffered async LDS fills; W pre-interleaved (pair layout).
// VOUT=true stores output in the attention-V pair-interleaved layout.
// ---------------------------------------------------------------------------
#define GBM 128
#define GBN 64
#define GBK 16
#define ALD 20   // GBK+4: rows 16B-aligned for b128 LDS writes; conflict-free

template <bool VOUT>
__global__ __launch_bounds__(256)
void gemm512_kernel(const float* __restrict__ A, const float* __restrict__ Wt,
                    const float* __restrict__ bias, float* __restrict__ C) {
  __shared__ float Als[2][GBM * ALD];      // 2 x 128x16 A tile (padded)
  __shared__ float Bls[2][(GBK / 2) * GBN * 2];  // 2 x pair-interleaved W tile

  const int tid  = threadIdx.x;
  const int wv   = tid >> 5;
  const int lane = tid & 31;
  const int hf   = lane >> 4;   // lane half
  const int l16  = lane & 15;
  const int nbase = blockIdx.x * GBN;
  const int mbase = blockIdx.y * GBM;

  // fixed per-thread copy assignments
  const int ar0 = tid >> 2,         ac0 = (tid & 3) * 4;   // A chunk 0
  const int ar1 = (tid + 256) >> 2, ac1 = ac0;             // A chunk 1
  const int bpr = tid >> 5;                                // W pair-row 0..7
  const int bcc = (tid & 31) * 4;                          // 0..124

  auto issue = [&](int kb, int buf) {
    copy16_g2l(A + (size_t)(mbase + ar0) * DMODEL + kb + ac0,
               &Als[buf][ar0 * ALD + ac0]);
    copy16_g2l(A + (size_t)(mbase + ar1) * DMODEL + kb + ac1,
               &Als[buf][ar1 * ALD + ac1]);
    // Wt pair-row (kb/2 + bpr): 1024 floats; need [nbase*2, nbase*2+128)
    copy16_g2l(Wt + (size_t)((kb >> 1) + bpr) * (DMODEL * 2) + nbase * 2 + bcc,
               &Bls[buf][bpr * (GBN * 2) + bcc]);
  };

  v8f acc[4];
  for (int nb = 0; nb < 4; ++nb) {
    const float bvv = bias[nbase + nb * 16 + l16];
    for (int r = 0; r < 8; ++r) acc[nb][r] = bvv;
  }

  issue(0, 0);
  for (int kb = 0; kb < DMODEL; kb += GBK) {
    const int buf = (kb >> 4) & 1;
    if (kb + GBK < DMODEL) {
      issue(kb + GBK, buf ^ 1);  // prefetch next tile into other buffer
      wait_async<3>();           // 3 just-issued pending -> current buf done
    } else {
      wait_async<0>();
    }
    __syncthreads();             // all waves' fills of current buf visible

    for (int kk = 0; kk < GBK; kk += 4) {
      // A frag: lane<16 -> A[m=l16][kk..kk+1]; lane>=16 -> [kk+2..kk+3]
      const v2f a =
          *(const v2f*)&Als[buf][(wv * 16 + l16) * ALD + kk + 2 * hf];
      const int p = (kk >> 1) + hf;  // k-pair index 0..7
      for (int nb = 0; nb < 4; ++nb) {
        // B frag: contiguous pair {B[k0][n], B[k0+1][n]} -> single b64
        const v2f b = *(const v2f*)&Bls[buf][(p * GBN + nb * 16 + l16) * 2];
        acc[nb] = wmma_f32(a, b, acc[nb]);
      }
    }
    __syncthreads();             // reads done before this buf is refilled
  }

  // C/D layout: vgpr r, lane<16 -> row r, col l16; lane>=16 -> row r+8
  for (int nb = 0; nb < 4; ++nb)
    for (int r = 0; r < 8; ++r) {
      const int row = mbase + wv * 16 + r + 8 * hf;
      const int col = nbase + nb * 16 + l16;
      if (VOUT) {
        // attention-V pair layout: key = row*8 + col/64, c = col%64;
        // addr = (key>>1)*128 + c*2 + (key&1)   (group-contiguous)
        const int j = col >> 6, c = col & 63;
        const size_t ka = (size_t)row * 8 + j;
        C[(ka >> 1) * 128 + c * 2 + (ka & 1)] = acc[nb][r];
      } else {
        C[(size_t)row * DMODEL + col] = acc[nb][r];
      }
    }
}

// ---------------------------------------------------------------------------
// Flash attention over contiguous [128, 1024, 64] Q/K (row-major) and
// V in pair-interleaved layout. block = 256 threads (8 waves) handles
// (group g, 128-query tile); online softmax over 64 key-tiles of 16.
// ---------------------------------------------------------------------------
#define KVLD 68  // DH+4: 16B-aligned rows, conflict-free (4*l16 mod 64)

__global__ __launch_bounds__(256)
void attn_kernel(const float* __restrict__ Q, const float* __restrict__ K,
                 const float* __restrict__ Vt, float* __restrict__ O) {
  __shared__ float Kls[2][16 * KVLD];    // K tile row-major, padded
  __shared__ float Vls[2][8 * DH * 2];   // V tile, pair-interleaved
  __shared__ float Pls[8 * 16 * 18];     // per-wave P tile (C->A bridge)

  const int tid  = threadIdx.x;
  const int wv   = tid >> 5;
  const int lane = tid & 31;
  const int hf   = lane >> 4;
  const int l16  = lane & 15;
  const int g     = blockIdx.x;
  const int qbase = blockIdx.y * 128;

  const float* Qg  = Q  + (size_t)g * SEQ * DH;
  const float* Kg  = K  + (size_t)g * SEQ * DH;
  const float* Vtg = Vt + (size_t)g * SEQ * DH;  // pair layout, same extent

  const int tr  = tid >> 4;        // K tile: key row, (tid&15)*4 = col chunk
  const int tc  = (tid & 15) * 4;
  const int vpr = tid >> 5;        // V tile: pair-row 0..7
  const int vcc = (tid & 31) * 4;  // 0..124

  auto issue = [&](int kt, int buf) {
    copy16_g2l(Kg + (size_t)(kt + tr) * DH + tc, &Kls[buf][tr * KVLD + tc]);
    copy16_g2l(Vtg + (size_t)((kt >> 1) + vpr) * (DH * 2) + vcc,
               &Vls[buf][vpr * (DH * 2) + vcc]);
  };

  // Q fragments for this wave's 16 rows, scaled by 1/sqrt(64); regs-resident.
  const int qrow = qbase + wv * 16 + l16;
  v2f qf[16];
  for (int p = 0; p < 16; ++p) {
    v2f t = *(const v2f*)(Qg + (size_t)qrow * DH + p * 4 + 2 * hf);
    t[0] *= 0.125f; t[1] *= 0.125f;
    qf[p] = t;
  }

  v8f oacc[4];
  for (int nb = 0; nb < 4; ++nb)
    for (int r = 0; r < 8; ++r) oacc[nb][r] = 0.f;
  float mrow[8], lrow[8];
  for (int r = 0; r < 8; ++r) { mrow[r] = -3.0e38f; lrow[r] = 0.f; }

  float* pw = &Pls[wv * 288];

  issue(0, 0);
  for (int kt = 0; kt < SEQ; kt += 16) {
    const int buf = (kt >> 4) & 1;
    if (kt + 16 < SEQ) {
      issue(kt + 16, buf ^ 1);
      wait_async<2>();
    } else {
      wait_async<0>();
    }
    __syncthreads();

    // S(16x16) = Qtile @ Ktile^T: B[c][j] = Ktile[j][c] -> contiguous b64
    v8f s;
    for (int r = 0; r < 8; ++r) s[r] = 0.f;
    for (int p = 0; p < 16; ++p) {
      const v2f b = *(const v2f*)&Kls[buf][l16 * KVLD + p * 4 + 2 * hf];
      s = wmma_f32(qf[p], b, s);
    }

    // online softmax: each output row lives in one 16-lane half of vgpr r
    float pv[8], alpha[8];
    for (int r = 0; r < 8; ++r) {
      float v = s[r];
      for (int off = 8; off >= 1; off >>= 1)
        v = fmaxf(v, __shfl_xor(v, off, 16));
      const float mnew = fmaxf(mrow[r], v);
      alpha[r] = __expf(mrow[r] - mnew);
      mrow[r]  = mnew;
      const float pe = __expf(s[r] - mnew);
      pv[r] = pe;
      float rs = pe;
      for (int off = 8; off >= 1; off >>= 1)
        rs += __shfl_xor(rs, off, 16);
      lrow[r] = lrow[r] * alpha[r] + rs;
    }
    for (int nb = 0; nb < 4; ++nb)
      for (int r = 0; r < 8; ++r) oacc[nb][r] *= alpha[r];

    // P tile: C-layout regs -> row-major LDS (wave-private; DS in-order)
    for (int r = 0; r < 8; ++r)
      pw[(r + 8 * hf) * 18 + l16] = pv[r];

    // O += P(16x16) @ Vtile(16x64); P as A frags, V frags = single b64
    for (int pc = 0; pc < 4; ++pc) {
      const v2f a = *(const v2f*)&pw[l16 * 18 + pc * 4 + 2 * hf];
      const int p = pc * 2 + hf;  // key-pair index 0..7
      for (int nb = 0; nb < 4; ++nb) {
        const v2f b = *(const v2f*)&Vls[buf][(p * DH + nb * 16 + l16) * 2];
        oacc[nb] = wmma_f32(a, b, oacc[nb]);
      }
    }
    __syncthreads();
  }

  float* Og = O + (size_t)g * SEQ * DH;
  for (int nb = 0; nb < 4; ++nb)
    for (int r = 0; r < 8; ++r)
      Og[(size_t)(qbase + wv * 16 + r + 8 * hf) * DH + nb * 16 + l16] =
          oacc[nb][r] / lrow[r];
}

// ---------------------------------------------------------------------------
extern "C" void kernel_launch(void* const* d_in, const int* in_sizes, int n_in,
                              void* d_out, int out_size, void* d_ws,
                              size_t ws_size, hipStream_t stream) {
  (void)in_sizes; (void)n_in; (void)out_size; (void)ws_size;
  const float* x   = (const float*)d_in[0];
  const float* wq  = (const float*)d_in[1];
  const float* bq  = (const float*)d_in[2];
  const float* wk  = (const float*)d_in[3];
  const float* bk  = (const float*)d_in[4];
  const float* wvp = (const float*)d_in[5];
  const float* bv  = (const float*)d_in[6];
  const float* wo  = (const float*)d_in[7];
  const float* bo  = (const float*)d_in[8];
  // d_in[9] = num_heads (8), baked into geometry above.

  float* Qb  = (float*)d_ws;                   // 32 MB
  float* Kb  = Qb + (size_t)NROW * DMODEL;     // 32 MB
  float* Vtb = Kb + (size_t)NROW * DMODEL;     // 32 MB (pair-interleaved)
  float* Wqt = Vtb + (size_t)NROW * DMODEL;    // 4 x 1 MB interleaved weights
  float* Wkt = Wqt + (size_t)DMODEL * DMODEL;
  float* Wvt = Wkt + (size_t)DMODEL * DMODEL;
  float* Wot = Wvt + (size_t)DMODEL * DMODEL;  // ws total ~= 100 MB
  float* AO  = Qb;  // alias: Q rows are register-resident in their owning wave
                    // before that wave writes the matching output rows.

  const int igrid = (DMODEL * DMODEL / 2) / 256;  // 512 blocks
  interleave_w_kernel<<<igrid, 256, 0, stream>>>(wq, Wqt);
  interleave_w_kernel<<<igrid, 256, 0, stream>>>(wk, Wkt);
  interleave_w_kernel<<<igrid, 256, 0, stream>>>(wvp, Wvt);
  interleave_w_kernel<<<igrid, 256, 0, stream>>>(wo, Wot);

  dim3 ggrid(DMODEL / GBN, NROW / GBM);  // (8, 128)
  gemm512_kernel<false><<<ggrid, 256, 0, stream>>>(x, Wqt, bq, Qb);
  gemm512_kernel<false><<<ggrid, 256, 0, stream>>>(x, Wkt, bk, Kb);
  gemm512_kernel<true ><<<ggrid, 256, 0, stream>>>(x, Wvt, bv, Vtb);
  attn_kernel<<<dim3(NGRP, SEQ / 128), 256, 0, stream>>>(Qb, Kb, Vtb, AO);
  gemm512_kernel<false><<<ggrid, 256, 0, stream>>>(AO, Wot, bo, (float*)d_out);
}